// GNN_MEAN_24249385353483
// MI455X (gfx1250) — compile-verified
//
#include <hip/hip_runtime.h>

typedef __attribute__((ext_vector_type(2))) float v2f;
typedef __attribute__((ext_vector_type(8))) float v8f;

// ---------------------------------------------------------------------------
// Degree count: one atomic add per edge on the destination node.
// ---------------------------------------------------------------------------
__global__ void gnn_deg_kernel(const int* __restrict__ ei, float* __restrict__ deg, int E) {
    int e = blockIdx.x * blockDim.x + threadIdx.x;
    if (e < E) {
        int d = ei[E + e];               // edge_index[1][e]
        atomicAdd(&deg[d], 1.0f);
    }
}

__global__ void gnn_invdeg_kernel(float* __restrict__ deg, int n) {
    int i = blockIdx.x * blockDim.x + threadIdx.x;
    if (i < n) deg[i] = 1.0f / fmaxf(deg[i], 1.0f);
}

// ---------------------------------------------------------------------------
// Scatter-add: agg[dst] += h[src].  grid.y indexes 4-wide feature chunks.
// ---------------------------------------------------------------------------
__global__ void gnn_scatter_kernel(const float* __restrict__ h, const int* __restrict__ ei,
                                   float* __restrict__ agg, int E, int cin, int chunk) {
    int e = blockIdx.x * blockDim.x + threadIdx.x;
    if (e >= E) return;
    int f = blockIdx.y * chunk;
    int s = ei[e];
    int d = ei[E + e];
    const float* hp = h + (size_t)s * cin + f;
    float*       ap = agg + (size_t)d * cin + f;
    if (chunk == 4) {
        float4 v = *reinterpret_cast<const float4*>(hp);
        atomicAdd(ap + 0, v.x);
        atomicAdd(ap + 1, v.y);
        atomicAdd(ap + 2, v.z);
        atomicAdd(ap + 3, v.w);
    } else {
        for (int j = 0; j < chunk; ++j) atomicAdd(ap + j, hp[j]);
    }
}

// ---------------------------------------------------------------------------
// SAGE layer GEMM: out = relu( (agg * inv_deg) @ Wl + bl + h @ Wr )
// One wave32 owns a 16-row tile and all COUT columns; fp32 WMMA 16x16x4.
// A layout (16x4 f32): lanes 0-15 hold K=k0,k0+1; lanes 16-31 hold K=k0+2,k0+3.
// C/D layout: VGPR r -> row m0+r (lanes 0-15) / m0+8+r (lanes 16-31), col = lane%16.
// Weights staged per-block into LDS as K-panels, pair-interleaved so each
// lane's two B elements are one ds_load_b64.
// ---------------------------------------------------------------------------
template <int CIN, int COUT>
__global__ __launch_bounds__(256) void gnn_sage_gemm_kernel(
    const float* __restrict__ agg, const float* __restrict__ h,
    const float* __restrict__ invdeg,
    const float* __restrict__ Wl, const float* __restrict__ bl,
    const float* __restrict__ Wr, float* __restrict__ out, int ntiles) {
    constexpr int NT = COUT / 16;
    constexpr int KP = (CIN >= 16) ? 16 : ((CIN + 3) & ~3);   // K-panel rows (padded)
    __shared__ float ldsW[2 * KP * COUT];                     // [Wl panel | Wr panel]

    int wave = blockIdx.x * 8 + (threadIdx.x >> 5);
    bool active = (wave < ntiles);              // wave-uniform
    int lane = threadIdx.x & 31;
    int half = lane >> 4;                       // 0: lanes 0-15, 1: lanes 16-31
    int l16  = lane & 15;
    int m0   = wave * 16;
    int rowA = m0 + l16;                        // A-matrix row owned by this lane
    float idg = 0.0f;
    if (active) idg = invdeg[rowA];

    v8f acc[NT] = {};

#pragma unroll 1
    for (int k0 = 0; k0 < CIN; k0 += KP) {
        __syncthreads();
        // Cooperative stage of Wl/Wr K-panel, pair-interleaved:
        //   lds[mtx*KP*COUT + (r>>1)*2*COUT + 2*c + (r&1)] = W[(k0+r)*COUT + c]
        // Zero-filled past CIN (layer 1 K-padding handled here, no masks later).
        constexpr int TOTAL = 2 * KP * COUT;
        for (int idx = threadIdx.x; idx < TOTAL; idx += 256) {
            int mtx = idx / (KP * COUT);
            int rem = idx - mtx * (KP * COUT);
            int r   = rem / COUT;
            int c   = rem - r * COUT;
            const float* W = mtx ? Wr : Wl;
            float v = 0.0f;
            if (k0 + r < CIN) v = W[(size_t)(k0 + r) * COUT + c];
            ldsW[mtx * (KP * COUT) + (r >> 1) * (2 * COUT) + 2 * c + (r & 1)] = v;
        }
        __syncthreads();

        if (active) {
#pragma unroll
            for (int ks = 0; ks < KP; ks += 4) {
                v2f aAgg, aH;
                if constexpr (CIN % 4 == 0) {
                    int ka = k0 + ks + 2 * half;             // this lane's first K elem
                    v2f a0 = *(const v2f*)&agg[(size_t)rowA * CIN + ka];
                    v2f h0 = *(const v2f*)&h[(size_t)rowA * CIN + ka];
                    aAgg[0] = a0[0] * idg; aAgg[1] = a0[1] * idg;
                    aH = h0;
                } else {
                    // CIN == 2: lanes 16-31 own K=2,3 which are zero padding.
                    float msk = (half == 0) ? 1.0f : 0.0f;
                    v2f a0 = *(const v2f*)&agg[(size_t)rowA * CIN];
                    v2f h0 = *(const v2f*)&h[(size_t)rowA * CIN];
                    aAgg[0] = a0[0] * idg * msk; aAgg[1] = a0[1] * idg * msk;
                    aH[0]   = h0[0] * msk;       aH[1]   = h0[1] * msk;
                }
                int pr = ((ks >> 1) + half) * (2 * COUT);    // pair row offset in LDS
#pragma unroll
                for (int t = 0; t < NT; ++t) {
                    int col = t * 16 + l16;
                    v2f bL = *(const v2f*)&ldsW[pr + 2 * col];
                    v2f bR = *(const v2f*)&ldsW[KP * COUT + pr + 2 * col];
                    acc[t] = __builtin_amdgcn_wmma_f32_16x16x4_f32(
                        false, aAgg, false, bL, (short)0, acc[t], false, false);
                    acc[t] = __builtin_amdgcn_wmma_f32_16x16x4_f32(
                        false, aH,   false, bR, (short)0, acc[t], false, false);
                }
            }
        }
    }

    if (active) {
#pragma unroll
        for (int t = 0; t < NT; ++t) {
            int col = t * 16 + l16;
            float bias = bl[col];
#pragma unroll
            for (int r = 0; r < 8; ++r) {
                int row = m0 + r + 8 * half;
                float v = acc[t][r] + bias;
                out[(size_t)row * COUT + col] = v > 0.0f ? v : 0.0f;
            }
        }
    }
}

// ---------------------------------------------------------------------------
// Final head: out[i] = h[i,:] @ W_out + b_out   (32 -> 1)
// ---------------------------------------------------------------------------
__global__ void gnn_out_kernel(const float* __restrict__ h, const float* __restrict__ Wout,
                               const float* __restrict__ bout, float* __restrict__ out, int n) {
    int i = blockIdx.x * blockDim.x + threadIdx.x;
    if (i >= n) return;
    float s = bout[0];
#pragma unroll
    for (int j = 0; j < 32; ++j) s += h[(size_t)i * 32 + j] * Wout[j];
    out[i] = s;
}

// ---------------------------------------------------------------------------
// Launch
// ---------------------------------------------------------------------------
extern "C" void kernel_launch(void* const* d_in, const int* in_sizes, int n_in,
                              void* d_out, int out_size, void* d_ws, size_t ws_size,
                              hipStream_t stream) {
    (void)n_in; (void)out_size; (void)ws_size;
    const float* x  = (const float*)d_in[0];
    const int*   ei = (const int*)d_in[1];
    const float* Wl[5] = {(const float*)d_in[3],  (const float*)d_in[6],  (const float*)d_in[9],
                          (const float*)d_in[12], (const float*)d_in[15]};
    const float* bl[5] = {(const float*)d_in[4],  (const float*)d_in[7],  (const float*)d_in[10],
                          (const float*)d_in[13], (const float*)d_in[16]};
    const float* Wr[5] = {(const float*)d_in[5],  (const float*)d_in[8],  (const float*)d_in[11],
                          (const float*)d_in[14], (const float*)d_in[17]};
    const float* Wout = (const float*)d_in[18];
    const float* bout = (const float*)d_in[19];
    float* outp = (float*)d_out;

    int n = in_sizes[0] / 2;       // 100000 nodes
    int E = in_sizes[1] / 2;       // 1600000 edges

    // Workspace: deg(n) | R0(n*128) | R1(n*128) | R2(n*128)
    float* deg = (float*)d_ws;
    float* R0  = deg + n;
    float* R1  = R0 + (size_t)n * 128;
    float* R2  = R1 + (size_t)n * 128;

    const int TB = 256;
    int eb = (E + TB - 1) / TB;
    int nb = (n + TB - 1) / TB;
    int ntiles  = (n + 15) / 16;           // 6250 (n is a multiple of 16)
    int gblocks = (ntiles + 7) / 8;        // 8 waves per block

    // in-degree -> inv_deg (in place, reused by every layer)
    hipMemsetAsync(deg, 0, (size_t)n * sizeof(float), stream);
    gnn_deg_kernel<<<eb, TB, 0, stream>>>(ei, deg, E);
    gnn_invdeg_kernel<<<nb, TB, 0, stream>>>(deg, n);

    auto scatter = [&](const float* hbuf, float* aggbuf, int cin) {
        hipMemsetAsync(aggbuf, 0, (size_t)n * cin * sizeof(float), stream);
        int chunk = cin >= 4 ? 4 : cin;
        dim3 grid(eb, cin / chunk);
        gnn_scatter_kernel<<<grid, TB, 0, stream>>>(hbuf, ei, aggbuf, E, cin, chunk);
    };

    // Layer 1: (2 -> 128)   h=x          agg->R0   out->R1
    scatter(x, R0, 2);
    gnn_sage_gemm_kernel<2, 128><<<gblocks, TB, 0, stream>>>(R0, x, deg, Wl[0], bl[0], Wr[0], R1, ntiles);
    // Layer 2: (128 -> 128) h=R1         agg->R2   out->R0
    scatter(R1, R2, 128);
    gnn_sage_gemm_kernel<128, 128><<<gblocks, TB, 0, stream>>>(R2, R1, deg, Wl[1], bl[1], Wr[1], R0, ntiles);
    // Layer 3: (128 -> 32)  h=R0         agg->R1   out->R2
    scatter(R0, R1, 128);
    gnn_sage_gemm_kernel<128, 32><<<gblocks, TB, 0, stream>>>(R1, R0, deg, Wl[2], bl[2], Wr[2], R2, ntiles);
    // Layer 4: (32 -> 32)   h=R2         agg->R0   out->R1
    scatter(R2, R0, 32);
    gnn_sage_gemm_kernel<32, 32><<<gblocks, TB, 0, stream>>>(R0, R2, deg, Wl[3], bl[3], Wr[3], R1, ntiles);
    // Layer 5: (32 -> 32)   h=R1         agg->R2   out->R0
    scatter(R1, R2, 32);
    gnn_sage_gemm_kernel<32, 32><<<gblocks, TB, 0, stream>>>(R2, R1, deg, Wl[4], bl[4], Wr[4], R0, ntiles);
    // Head: (32 -> 1)
    gnn_out_kernel<<<nb, TB, 0, stream>>>(R0, Wout, bout, outp, n);
}